// GATLayer_21964462752227
// MI455X (gfx1250) — compile-verified
//
#include <hip/hip_runtime.h>
#include <hip/hip_bf16.h>
#include <math.h>

typedef __attribute__((ext_vector_type(2))) float v2f;
typedef __attribute__((ext_vector_type(8))) float v8f;

#define F_IN   64
#define F_OUT  32
#define NEG_SLOPE 0.2f

// ---------------------------------------------------------------------------
// Pass 0: initialize m = -inf, denom = 0, out = 0
// ---------------------------------------------------------------------------
__global__ __launch_bounds__(256) void gat_init(float* __restrict__ m,
                                                float* __restrict__ denom,
                                                float* __restrict__ out,
                                                int n_nodes, int total_out) {
    int i = blockIdx.x * blockDim.x + threadIdx.x;
    if (i < n_nodes) { m[i] = -INFINITY; denom[i] = 0.0f; }
    if (i < total_out) out[i] = 0.0f;
}

// ---------------------------------------------------------------------------
// Pass 1: h = x @ W  via V_WMMA_F32_16X16X4_F32 (full fp32 fidelity).
// One wave computes a 16x16 tile of h; K=64 consumed in 16 steps of 4.
// Block = 256 threads = 8 waves = 4 row-tiles x 2 col-tiles.
// ---------------------------------------------------------------------------
__global__ __launch_bounds__(256) void gat_gemm_wmma(const float* __restrict__ x,
                                                     const float* __restrict__ W,
                                                     float* __restrict__ h,
                                                     int n_nodes) {
    const int wave  = threadIdx.x >> 5;          // 0..7
    const int lane  = threadIdx.x & 31;
    const int tile  = blockIdx.x * 4 + (wave >> 1);   // row-tile index
    const int colt  = wave & 1;                       // col-tile 0/1
    const int row0  = tile * 16;
    if (row0 >= n_nodes) return;                      // wave-uniform guard
    const int col0  = colt * 16;
    const int r     = lane & 15;
    const int khalf = (lane >> 4) * 2;                // 0 for lanes 0-15, 2 for 16-31

    v8f c = {};
    #pragma unroll
    for (int k = 0; k < F_IN; k += 4) {
        // A: 16x4 tile of x. lane<16: K = k,k+1 ; lane>=16: K = k+2,k+3
        const float* xp = x + (size_t)(row0 + r) * F_IN + k + khalf;
        v2f a; a.x = xp[0]; a.y = xp[1];
        // B: 4x16 tile of W (row K, col N). VGPR0: K=k|k+2, VGPR1: K=k+1|k+3
        const float* wp = W + (size_t)(k + khalf) * F_OUT + col0 + r;
        v2f b; b.x = wp[0]; b.y = wp[F_OUT];
        c = __builtin_amdgcn_wmma_f32_16x16x4_f32(
                /*neg_a=*/false, a, /*neg_b=*/false, b,
                /*c_mod=*/(short)0, c, /*reuse_a=*/false, /*reuse_b=*/false);
    }
    // D layout: VGPR v -> row = v + 8*(lane>=16), col = lane&15
    const int mrow = (lane >> 4) * 8;
    #pragma unroll
    for (int v = 0; v < 8; ++v) {
        h[(size_t)(row0 + mrow + v) * F_OUT + col0 + r] = c[v];
    }
}

// ---------------------------------------------------------------------------
// Pass 2: per-node scalar projections s = h . a_src, t = h . a_dst
// One wave (32 lanes) per node; lane = feature; wave32 shuffle reduction.
// ---------------------------------------------------------------------------
__global__ __launch_bounds__(256) void gat_proj(const float* __restrict__ h,
                                                const float* __restrict__ a_src,
                                                const float* __restrict__ a_dst,
                                                float* __restrict__ s,
                                                float* __restrict__ t,
                                                int n_nodes) {
    int node = (blockIdx.x * blockDim.x + threadIdx.x) >> 5;
    int lane = threadIdx.x & 31;
    if (node >= n_nodes) return;
    float hv = h[(size_t)node * F_OUT + lane];
    float ps = hv * a_src[lane];
    float pt = hv * a_dst[lane];
    #pragma unroll
    for (int off = 16; off > 0; off >>= 1) {
        ps += __shfl_down(ps, off, 32);
        pt += __shfl_down(pt, off, 32);
    }
    if (lane == 0) { s[node] = ps; t[node] = pt; }
}

// ---------------------------------------------------------------------------
// Pass 3: e = leaky_relu(s[src]+t[dst]); segment-max into m via sign-split
// integer atomics (lowers to GLOBAL_ATOMIC_MAX_I32 / MIN_U32, L2-resolved).
// ---------------------------------------------------------------------------
__global__ __launch_bounds__(256) void gat_edge_logits(const int* __restrict__ src,
                                                       const int* __restrict__ dst,
                                                       const float* __restrict__ s,
                                                       const float* __restrict__ t,
                                                       float* __restrict__ e,
                                                       float* __restrict__ m,
                                                       int n_edges) {
    int i = blockIdx.x * blockDim.x + threadIdx.x;
    if (i >= n_edges) return;
    float v = s[src[i]] + t[dst[i]];
    v = (v > 0.0f) ? v : NEG_SLOPE * v;
    e[i] = v;
    int d = dst[i];
    if (v >= 0.0f) atomicMax((int*)(m + d), __float_as_int(v));
    else           atomicMin((unsigned int*)(m + d), __float_as_uint(v));
}

// ---------------------------------------------------------------------------
// Pass 4: ex = exp(e - m[dst]) (stored over e); denom[dst] += ex
// ---------------------------------------------------------------------------
__global__ __launch_bounds__(256) void gat_edge_exp(const int* __restrict__ dst,
                                                    float* __restrict__ e,
                                                    const float* __restrict__ m,
                                                    float* __restrict__ denom,
                                                    int n_edges) {
    int i = blockIdx.x * blockDim.x + threadIdx.x;
    if (i >= n_edges) return;
    int d = dst[i];
    float ex = __expf(e[i] - m[d]);
    e[i] = ex;
    atomicAdd(denom + d, ex);
}

// ---------------------------------------------------------------------------
// Pass 5: out[dst] += alpha * h[src]. One wave per edge, lane = feature.
// Coalesced 128B gather of h[src] (L2-resident) + 32 native f32 atomics.
// ---------------------------------------------------------------------------
__global__ __launch_bounds__(256) void gat_scatter(const int* __restrict__ src,
                                                   const int* __restrict__ dst,
                                                   const float* __restrict__ ex,
                                                   const float* __restrict__ denom,
                                                   const float* __restrict__ h,
                                                   float* __restrict__ out,
                                                   int n_edges) {
    int edge = (blockIdx.x * blockDim.x + threadIdx.x) >> 5;
    int lane = threadIdx.x & 31;
    if (edge >= n_edges) return;
    int sN = src[edge], dN = dst[edge];
    float alpha = ex[edge] / (denom[dN] + 1e-16f);
    float val = alpha * h[(size_t)sN * F_OUT + lane];
    atomicAdd(out + (size_t)dN * F_OUT + lane, val);
}

// ---------------------------------------------------------------------------
// Pass 6: ELU in place on out
// ---------------------------------------------------------------------------
__global__ __launch_bounds__(256) void gat_elu(float* __restrict__ out, int total) {
    int i = blockIdx.x * blockDim.x + threadIdx.x;
    if (i >= total) return;
    float v = out[i];
    out[i] = (v > 0.0f) ? v : (__expf(v) - 1.0f);
}

// ---------------------------------------------------------------------------
extern "C" void kernel_launch(void* const* d_in, const int* in_sizes, int n_in,
                              void* d_out, int out_size, void* d_ws, size_t ws_size,
                              hipStream_t stream) {
    const float* x     = (const float*)d_in[0];
    const int*   eidx  = (const int*)  d_in[1];
    const float* W     = (const float*)d_in[2];
    const float* a_src = (const float*)d_in[3];
    const float* a_dst = (const float*)d_in[4];
    float*       out   = (float*)d_out;

    const int n_nodes = in_sizes[0] / F_IN;     // 100000
    const int n_edges = in_sizes[1] / 2;        // 1600000
    const int* src = eidx;
    const int* dst = eidx + n_edges;

    // Workspace partition (floats): h[N*32] | s[N] | t[N] | m[N] | denom[N] | e[E]
    float* h     = (float*)d_ws;
    float* s     = h + (size_t)n_nodes * F_OUT;
    float* t     = s + n_nodes;
    float* m     = t + n_nodes;
    float* denom = m + n_nodes;
    float* e     = denom + n_nodes;

    const int total_out = n_nodes * F_OUT;

    // Pass 0: init
    gat_init<<<(total_out + 255) / 256, 256, 0, stream>>>(m, denom, out, n_nodes, total_out);

    // Pass 1: h = x @ W  (WMMA f32)
    int row_tiles = (n_nodes + 15) / 16;        // 6250 (exact)
    gat_gemm_wmma<<<(row_tiles + 3) / 4, 256, 0, stream>>>(x, W, h, n_nodes);

    // Pass 2: per-node projections
    gat_proj<<<((size_t)n_nodes * 32 + 255) / 256, 256, 0, stream>>>(h, a_src, a_dst, s, t, n_nodes);

    // Pass 3: logits + segment max
    gat_edge_logits<<<(n_edges + 255) / 256, 256, 0, stream>>>(src, dst, s, t, e, m, n_edges);

    // Pass 4: exp + segment sum
    gat_edge_exp<<<(n_edges + 255) / 256, 256, 0, stream>>>(dst, e, m, denom, n_edges);

    // Pass 5: weighted scatter-add (wave per edge)
    gat_scatter<<<((size_t)n_edges * 32 + 255) / 256, 256, 0, stream>>>(src, dst, e, denom, h, out, n_edges);

    // Pass 6: ELU
    gat_elu<<<(total_out + 255) / 256, 256, 0, stream>>>(out, total_out);
}